// TransitionUp_18923625906663
// MI455X (gfx1250) — compile-verified
//
#include <hip/hip_runtime.h>
#include <hip/hip_bf16.h>

typedef __attribute__((ext_vector_type(2))) float v2f;
typedef __attribute__((ext_vector_type(8))) float v8f;

#define BN   2
#define N1D  4096
#define N2D  16384
#define CIN  512
#define CSK  256
#define COUT 256

// B tile LDS layout: K-pair interleaved. Element (k, n) of the 64x128 tile at
// lsB[(k>>1)*BSTRIDE + n*2 + (k&1)]  -> a WMMA B fragment (K=koff, koff+1 at
// column n) is ONE aligned ds_load_b64. 32 k-pairs, row stride 258 (pad 2).
#define BSTRIDE 258

// ---------------------------------------------------------------------------
// Fused GEMM + BatchNorm + ReLU using fp32 WMMA (V_WMMA_F32_16X16X4_F32).
// out[o,n] = relu(scale[o] * sum_c W[o,c]*X[b,c,n] + shift[o])
// Macro tile: 64 (o) x 128 (n), K blocked by 64.
// A tile staged via CDNA5 GLOBAL_LOAD_ASYNC_TO_LDS_B128 (ASYNCcnt),
// B tile staged pair-interleaved through VGPRs.
// 8 waves: wave w -> o-subtile (w>>1), n-half (w&1) covering 4x 16x16 tiles.
// layoutNC==1 : out is (B, Ndim, COUT)  (channel-contiguous, for gather)
// layoutNC==0 : out is (B, COUT, Ndim)  (final y layout)
// ---------------------------------------------------------------------------
__global__ __launch_bounds__(256)
void gemm_bn_relu(const float* __restrict__ W,
                  const float* __restrict__ X,
                  const float* __restrict__ gam, const float* __restrict__ bet,
                  const float* __restrict__ mu,  const float* __restrict__ var,
                  float* __restrict__ out,
                  int Kdim, int Ndim, int layoutNC)
{
    __shared__ float lsA[64 * 66];        // 64 o x 64 k, padded row stride 66
    __shared__ float lsB[32 * BSTRIDE];   // 32 k-pairs x 128 n interleaved
    __shared__ float sSc[64], sSh[64];

    const int tid  = threadIdx.x;
    const int lane = tid & 31;
    const int wv   = tid >> 5;        // wave 0..7
    const int g    = lane >> 4;       // lane half 0/1
    const int l16  = lane & 15;

    const int nBase = blockIdx.x * 128;
    const int oBase = blockIdx.y * 64;
    const int bi    = blockIdx.z;

    if (tid < 64) {
        const int o = oBase + tid;
        const float s = gam[o] * rsqrtf(var[o] + 1e-5f);
        sSc[tid] = s;
        sSh[tid] = bet[o] - mu[o] * s;
    }

    const float* __restrict__ Xb = X + (size_t)bi * Kdim * Ndim;

    v8f acc[4];
#pragma unroll
    for (int i = 0; i < 4; ++i)
#pragma unroll
        for (int j = 0; j < 8; ++j) acc[i][j] = 0.0f;

    const int oSub  = wv >> 1;        // 0..3
    const int nHalf = wv & 1;         // 0..1
    const int aRow  = oSub * 16 + l16;

    for (int k0 = 0; k0 < Kdim; k0 += 64) {
        __syncthreads();
        // ---- stage A tile: 64 o x 64 k, async global->LDS, 16B per lane/iter
#pragma unroll
        for (int i = 0; i < 4; ++i) {
            const int s  = tid + i * 256;
            const int ar = s >> 4;        // o row 0..63
            const int aq = s & 15;        // float4 column
            const float* gp = W + (size_t)(oBase + ar) * Kdim + k0 + aq * 4;
            // low 32 bits of a generic LDS pointer == LDS byte offset
            const unsigned ldsOff = (unsigned)(size_t)(&lsA[ar * 66 + aq * 4]);
            asm volatile("global_load_async_to_lds_b128 %0, %1, off"
                         :
                         : "v"(ldsOff), "v"((unsigned long long)(size_t)gp)
                         : "memory");
        }
        // ---- stage B tile: 64 k x 128 n, pair-interleaved (2048 float4 / 256 thr)
#pragma unroll
        for (int i = 0; i < 8; ++i) {
            const int s  = tid + i * 256;
            const int kr = s >> 5;        // k row 0..63
            const int nq = s & 31;        // float4 column
            const float4 bv = *(const float4*)(Xb + (size_t)(k0 + kr) * Ndim + nBase + nq * 4);
            float* dst = &lsB[(kr >> 1) * BSTRIDE + nq * 8 + (kr & 1)];
            dst[0] = bv.x; dst[2] = bv.y; dst[4] = bv.z; dst[6] = bv.w;
        }
        asm volatile("s_wait_asynccnt 0x0" ::: "memory");
        __syncthreads();

        // ---- compute: 16 K-steps of 4, fp32 WMMA
#pragma unroll 4
        for (int kk = 0; kk < 64; kk += 4) {
            const int koff  = kk + 2 * g;                 // even; K = koff + vgpr idx
            const int kpair = (kk >> 1) + g;
            v2f a = *(const v2f*)&lsA[aRow * 66 + koff];  // ds_load_b64
#pragma unroll
            for (int nt = 0; nt < 4; ++nt) {
                const int col = nHalf * 64 + nt * 16 + l16;
                v2f b = *(const v2f*)&lsB[kpair * BSTRIDE + col * 2]; // ds_load_b64
                acc[nt] = __builtin_amdgcn_wmma_f32_16x16x4_f32(
                    false, a, false, b, (short)0, acc[nt], false, false);
            }
        }
    }

    // ---- epilogue: BN + ReLU, store
#pragma unroll
    for (int nt = 0; nt < 4; ++nt) {
        const int n    = nBase + nHalf * 64 + nt * 16 + l16;
        const int oLoc = oSub * 16 + 8 * g;   // 8 consecutive local o (D-layout: M = r + 8g)
        if (layoutNC) {
            float* dst = out + ((size_t)bi * Ndim + n) * COUT + oBase + oLoc;
#pragma unroll
            for (int r = 0; r < 8; ++r) {
                const float vv = acc[nt][r] * sSc[oLoc + r] + sSh[oLoc + r];
                dst[r] = vv > 0.0f ? vv : 0.0f;   // merged into b128 stores
            }
        } else {
#pragma unroll
            for (int r = 0; r < 8; ++r) {
                const int o = oBase + oLoc + r;
                const float vv = acc[nt][r] * sSc[oLoc + r] + sSh[oLoc + r];
                out[((size_t)bi * COUT + o) * Ndim + n] = vv > 0.0f ? vv : 0.0f;
            }
        }
    }
}

// ---------------------------------------------------------------------------
// 3-NN search: one thread per p2 point, p1 staged in LDS tiles of 256.
// Branchless running top-3 (smallest squared distances). Also copies p2 -> out.
// ---------------------------------------------------------------------------
__global__ __launch_bounds__(256)
void knn3(const float* __restrict__ p1, const float* __restrict__ p2,
          int* __restrict__ idxOut, float* __restrict__ wOut,
          float* __restrict__ p2Out)
{
    __shared__ float sx[256], sy[256], sz[256];

    const int gid = blockIdx.x * 256 + threadIdx.x;
    const int b   = gid / N2D;          // N2D % 256 == 0 -> uniform per block
    const int n2  = gid % N2D;

    const size_t pbase = ((size_t)b * N2D + n2) * 3;
    const float px = p2[pbase + 0];
    const float py = p2[pbase + 1];
    const float pz = p2[pbase + 2];
    p2Out[pbase + 0] = px;
    p2Out[pbase + 1] = py;
    p2Out[pbase + 2] = pz;

    float d0 = 3.4e38f, d1 = 3.4e38f, d2 = 3.4e38f;
    int   i0 = 0, i1 = 0, i2 = 0;

    for (int t0 = 0; t0 < N1D; t0 += 256) {
        __syncthreads();
        {
            const int j = t0 + threadIdx.x;
            const size_t q = ((size_t)b * N1D + j) * 3;
            sx[threadIdx.x] = p1[q + 0];
            sy[threadIdx.x] = p1[q + 1];
            sz[threadIdx.x] = p1[q + 2];
        }
        __syncthreads();
#pragma unroll 4
        for (int j = 0; j < 256; ++j) {
            const float dx = px - sx[j];
            const float dy = py - sy[j];
            const float dz = pz - sz[j];
            const float d  = dx * dx + dy * dy + dz * dz;
            const int  id  = t0 + j;
            const bool lt0 = d < d0, lt1 = d < d1, lt2 = d < d2;
            d2 = lt1 ? d1 : (lt2 ? d  : d2);  i2 = lt1 ? i1 : (lt2 ? id : i2);
            d1 = lt0 ? d0 : (lt1 ? d  : d1);  i1 = lt0 ? i0 : (lt1 ? id : i1);
            d0 = lt0 ? d  : d0;               i0 = lt0 ? id : i0;
        }
    }

    const float r0 = 1.0f / (d0 + 1e-8f);
    const float r1 = 1.0f / (d1 + 1e-8f);
    const float r2 = 1.0f / (d2 + 1e-8f);
    const float rs = 1.0f / (r0 + r1 + r2);
    const size_t o = ((size_t)b * N2D + n2) * 3;
    idxOut[o + 0] = i0;  wOut[o + 0] = r0 * rs;
    idxOut[o + 1] = i1;  wOut[o + 1] = r1 * rs;
    idxOut[o + 2] = i2;  wOut[o + 2] = r2 * rs;
}

// ---------------------------------------------------------------------------
// y(B,C,N2) += sum_k w_k * h1_nc(B,N1,C)[idx_k]
// Block = 32 n2 points x 256 channels. Phase A: coalesced channel-contiguous
// gathers into padded LDS (stride 257 -> conflict-free transposed reads).
// Phase B: coalesced += along n2 in the (C,N2) output layout.
// ---------------------------------------------------------------------------
__global__ __launch_bounds__(256)
void upsample_add(const float* __restrict__ h1,
                  const int* __restrict__ idxIn, const float* __restrict__ wIn,
                  float* __restrict__ y)
{
    __shared__ float lsU[32 * 257];
    __shared__ int   sIdx[32 * 3];
    __shared__ float sW[32 * 3];

    const int b      = blockIdx.y;
    const int n2Base = blockIdx.x * 32;
    const int tid    = threadIdx.x;

    if (tid < 96) {
        const size_t o = ((size_t)b * N2D + n2Base) * 3 + tid;
        sIdx[tid] = idxIn[o];
        sW[tid]   = wIn[o];
    }
    __syncthreads();

    const float* __restrict__ h1b = h1 + (size_t)b * N1D * COUT;
    const int c = tid;  // channel 0..255
#pragma unroll 2
    for (int nl = 0; nl < 32; ++nl) {
        const int   j0 = sIdx[nl * 3 + 0], j1 = sIdx[nl * 3 + 1], j2 = sIdx[nl * 3 + 2];
        const float w0 = sW[nl * 3 + 0],  w1 = sW[nl * 3 + 1],  w2 = sW[nl * 3 + 2];
        lsU[nl * 257 + c] = w0 * h1b[(size_t)j0 * COUT + c]
                          + w1 * h1b[(size_t)j1 * COUT + c]
                          + w2 * h1b[(size_t)j2 * COUT + c];
    }
    __syncthreads();

    const int nl = tid & 31;       // lanes 0..31 -> consecutive n2 (coalesced)
    const int cq = tid >> 5;       // 0..7
#pragma unroll 4
    for (int cb = 0; cb < 32; ++cb) {
        const int cc = cb * 8 + cq;
        float* yp = y + ((size_t)b * COUT + cc) * N2D + n2Base + nl;
        *yp += lsU[nl * 257 + cc];
    }
}

// ---------------------------------------------------------------------------
extern "C" void kernel_launch(void* const* d_in, const int* in_sizes, int n_in,
                              void* d_out, int out_size, void* d_ws, size_t ws_size,
                              hipStream_t stream)
{
    (void)in_sizes; (void)n_in; (void)out_size; (void)ws_size;

    const float* p1 = (const float*)d_in[0];
    const float* x1 = (const float*)d_in[1];
    const float* p2 = (const float*)d_in[2];
    const float* x2 = (const float*)d_in[3];
    const float* W1 = (const float*)d_in[4];
    const float* g1 = (const float*)d_in[5];
    const float* b1 = (const float*)d_in[6];
    const float* m1 = (const float*)d_in[7];
    const float* v1 = (const float*)d_in[8];
    const float* W2 = (const float*)d_in[9];
    const float* g2 = (const float*)d_in[10];
    const float* b2 = (const float*)d_in[11];
    const float* m2 = (const float*)d_in[12];
    const float* v2 = (const float*)d_in[13];

    float* p2Out = (float*)d_out;                                  // B*N2*3
    float* yOut  = (float*)d_out + (size_t)BN * N2D * 3;           // B*COUT*N2

    // workspace: h1_nc (8 MB) | idx (384 KB) | w (384 KB)
    float* h1   = (float*)d_ws;
    int*   idxW = (int*)((char*)d_ws + (size_t)BN * N1D * COUT * sizeof(float));
    float* wW   = (float*)((char*)idxW + (size_t)BN * N2D * 3 * sizeof(int));

    // GEMM1: h1 = relu(bn(W1 @ x1)), stored (B, N1, C) for coalesced gathers
    gemm_bn_relu<<<dim3(N1D / 128, COUT / 64, BN), 256, 0, stream>>>(
        W1, x1, g1, b1, m1, v1, h1, CIN, N1D, 1);

    // GEMM2: y = relu(bn(W2 @ x2)) in final (B, C, N2) layout
    gemm_bn_relu<<<dim3(N2D / 128, COUT / 64, BN), 256, 0, stream>>>(
        W2, x2, g2, b2, m2, v2, yOut, CSK, N2D, 0);

    // 3-NN + weights + p2 passthrough
    knn3<<<(BN * N2D) / 256, 256, 0, stream>>>(p1, p2, idxW, wW, p2Out);

    // y += interpolated h1
    upsample_add<<<dim3(N2D / 32, BN), 256, 0, stream>>>(h1, idxW, wW, yOut);
}